// Relation_44255343018632
// MI455X (gfx1250) — compile-verified
//
#include <hip/hip_runtime.h>

typedef __attribute__((ext_vector_type(2))) float v2f;
typedef __attribute__((ext_vector_type(4))) float v4f;
typedef __attribute__((ext_vector_type(8))) float v8f;

constexpr int B_ = 8, N_ = 1024, F_ = 512, R_ = 32;
constexpr int ROWS_PER_WAVE   = 16;
constexpr int WAVES_PER_BLOCK = 8;
constexpr int ROWS_PER_BLOCK  = ROWS_PER_WAVE * WAVES_PER_BLOCK;   // 128
constexpr int BLOCKS_PER_BATCH = N_ / ROWS_PER_BLOCK;              // 8
constexpr int NUM_GEMM_BLOCKS  = B_ * BLOCKS_PER_BATCH;            // 64

// workspace layout (in floats)
constexpr size_t WS_WT   = 0;                                      // 3*R_*F_  transposed weights Wt[m][r][k]
constexpr size_t WS_PART = WS_WT + (size_t)3 * R_ * F_;            // NUM_GEMM_BLOCKS*2*R_
constexpr size_t WS_GSUM = WS_PART + (size_t)NUM_GEMM_BLOCKS * 2 * R_;
constexpr size_t WS_CSUM = WS_GSUM + (size_t)B_ * R_;

// ---------------------------------------------------------------------------
// K0: transpose the three [F,R] weight matrices into Wt[m][r][k] ([3][32][512])
// so B-matrix WMMA fragments become single b64 loads per lane.
__global__ void transpose_w_kernel(const float* __restrict__ W1,
                                   const float* __restrict__ W2,
                                   const float* __restrict__ Wg,
                                   float* __restrict__ wt) {
    int idx = blockIdx.x * blockDim.x + threadIdx.x;   // 0 .. 3*F_*R_-1
    int m   = idx / (F_ * R_);
    int rem = idx - m * (F_ * R_);
    int k   = rem / R_;
    int r   = rem - k * R_;
    const float* W = (m == 0) ? W1 : ((m == 1) ? W2 : Wg);
    wt[((size_t)m * R_ + r) * F_ + k] = W[rem];        // coalesced read, scattered write
}

// ---------------------------------------------------------------------------
// K1: the three fused GEMMs via V_WMMA_F32_16X16X4_F32.
// Each wave: 16 rows x 32 cols x 3 matrices = 6 f32 accumulators.
// Writes z1 (=x@W1, no bias) into out, per-block partial sums of g and z2*g
// into ws (fixed-order tree -> deterministic, no float atomics).
__global__ void __launch_bounds__(256)
gemm3_kernel(const float* __restrict__ x, const float* __restrict__ wt,
             float* __restrict__ z1_out, float* __restrict__ partials) {
    const int wave   = threadIdx.x >> 5;
    const int lane   = threadIdx.x & 31;
    const int lane16 = lane & 15;
    const int laneHi = lane >> 4;         // 0: K pair {0,1}, 1: K pair {2,3}
    const int batch  = blockIdx.x >> 3;
    const int blkInB = blockIdx.x & 7;
    const int rowBase = blkInB * ROWS_PER_BLOCK + wave * ROWS_PER_WAVE;

    // A fragment source: lane16 -> row M, laneHi selects k pair.
    const float* ax = x + ((size_t)(batch * N_ + rowBase + lane16)) * F_ + 2 * laneHi;
    // B fragment sources (transposed weights): lane16 -> col N (= r), laneHi -> k pair.
    const float* bw00 = wt + ((size_t)(0 * R_ +  0 + lane16)) * F_ + 2 * laneHi; // W1, r tile 0
    const float* bw01 = wt + ((size_t)(0 * R_ + 16 + lane16)) * F_ + 2 * laneHi; // W1, r tile 1
    const float* bw10 = wt + ((size_t)(1 * R_ +  0 + lane16)) * F_ + 2 * laneHi; // W2
    const float* bw11 = wt + ((size_t)(1 * R_ + 16 + lane16)) * F_ + 2 * laneHi;
    const float* bw20 = wt + ((size_t)(2 * R_ +  0 + lane16)) * F_ + 2 * laneHi; // Wg
    const float* bw21 = wt + ((size_t)(2 * R_ + 16 + lane16)) * F_ + 2 * laneHi;

    v8f a00 = {}, a01 = {};   // z1 accumulators (tiles r0, r1)
    v8f a10 = {}, a11 = {};   // z2
    v8f a20 = {}, a21 = {};   // g

#pragma unroll 4
    for (int k0 = 0; k0 < F_; k0 += 4) {
        v2f a   = *(const v2f*)(ax   + k0);
        v2f b00 = *(const v2f*)(bw00 + k0);
        v2f b01 = *(const v2f*)(bw01 + k0);
        v2f b10 = *(const v2f*)(bw10 + k0);
        v2f b11 = *(const v2f*)(bw11 + k0);
        v2f b20 = *(const v2f*)(bw20 + k0);
        v2f b21 = *(const v2f*)(bw21 + k0);
        a00 = __builtin_amdgcn_wmma_f32_16x16x4_f32(false, a, false, b00, (short)0, a00, false, false);
        a01 = __builtin_amdgcn_wmma_f32_16x16x4_f32(false, a, false, b01, (short)0, a01, false, false);
        a10 = __builtin_amdgcn_wmma_f32_16x16x4_f32(false, a, false, b10, (short)0, a10, false, false);
        a11 = __builtin_amdgcn_wmma_f32_16x16x4_f32(false, a, false, b11, (short)0, a11, false, false);
        a20 = __builtin_amdgcn_wmma_f32_16x16x4_f32(false, a, false, b20, (short)0, a20, false, false);
        a21 = __builtin_amdgcn_wmma_f32_16x16x4_f32(false, a, false, b21, (short)0, a21, false, false);
    }

    // Store z1 tiles. C/D layout: VGPR v, lanes 0-15 -> row v, lanes 16-31 -> row v+8; col = lane16.
    float* outp = z1_out + ((size_t)(batch * N_ + rowBase + laneHi * 8)) * R_ + lane16;
#pragma unroll
    for (int v = 0; v < 8; ++v) {
        outp[(size_t)v * R_]      = a00[v];
        outp[(size_t)v * R_ + 16] = a01[v];
    }

    // Per-lane partial sums over this lane's 8 rows: g and z2*g.
    float g0 = 0.f, g1 = 0.f, zg0 = 0.f, zg1 = 0.f;
#pragma unroll
    for (int v = 0; v < 8; ++v) {
        g0  += a20[v];
        g1  += a21[v];
        zg0 += a10[v] * a20[v];
        zg1 += a11[v] * a21[v];
    }

    __shared__ float red[2][WAVES_PER_BLOCK][R_][2];
    red[0][wave][lane16     ][laneHi] = g0;
    red[0][wave][lane16 + 16][laneHi] = g1;
    red[1][wave][lane16     ][laneHi] = zg0;
    red[1][wave][lane16 + 16][laneHi] = zg1;
    __syncthreads();

    if (threadIdx.x < 64) {
        int s = threadIdx.x >> 5;     // 0 = g_sum, 1 = z2g_sum
        int r = threadIdx.x & 31;
        float p = 0.f;
#pragma unroll
        for (int w = 0; w < WAVES_PER_BLOCK; ++w) {
            p += red[s][w][r][0];
            p += red[s][w][r][1];
        }
        partials[((size_t)blockIdx.x * 2 + s) * R_ + r] = p;
    }
}

// ---------------------------------------------------------------------------
// K2: reduce per-block partials -> per-batch gsum and csum = bias*gsum + z2g_sum.
__global__ void reduce_kernel(const float* __restrict__ partials,
                              const float* __restrict__ bias,
                              float* __restrict__ gsum, float* __restrict__ csum) {
    int tid = threadIdx.x;            // 256 = B_ * R_
    int b = tid >> 5;
    int r = tid & 31;
    float g = 0.f, zg = 0.f;
#pragma unroll
    for (int blk = 0; blk < BLOCKS_PER_BATCH; ++blk) {
        size_t base = ((size_t)(b * BLOCKS_PER_BATCH + blk) * 2) * R_ + r;
        g  += partials[base];
        zg += partials[base + R_];
    }
    gsum[b * R_ + r] = g;
    csum[b * R_ + r] = bias[r] * g + zg;
}

// ---------------------------------------------------------------------------
// K3: out = z1 * gsum[b] + csum[b], float4 vectorized in place.
__global__ void epilogue_kernel(float* __restrict__ out,
                                const float* __restrict__ gsum,
                                const float* __restrict__ csum) {
    int i = blockIdx.x * blockDim.x + threadIdx.x;   // float4 index
    size_t base = (size_t)i << 2;
    int b = (int)(base >> 15);                       // / (N_*R_) = 32768
    int r = (int)(base & (R_ - 1));                  // multiple of 4
    v4f v = *(v4f*)(out + base);
    v4f g = *(const v4f*)(gsum + b * R_ + r);
    v4f c = *(const v4f*)(csum + b * R_ + r);
    *(v4f*)(out + base) = v * g + c;
}

// ---------------------------------------------------------------------------
extern "C" void kernel_launch(void* const* d_in, const int* in_sizes, int n_in,
                              void* d_out, int out_size, void* d_ws, size_t ws_size,
                              hipStream_t stream) {
    (void)in_sizes; (void)n_in; (void)out_size; (void)ws_size;
    const float* x    = (const float*)d_in[0];
    const float* W1   = (const float*)d_in[1];
    const float* W2   = (const float*)d_in[2];
    const float* Wg   = (const float*)d_in[3];
    const float* bias = (const float*)d_in[4];
    float* out = (float*)d_out;
    float* ws  = (float*)d_ws;

    float* wt       = ws + WS_WT;
    float* partials = ws + WS_PART;
    float* gsum     = ws + WS_GSUM;
    float* csum     = ws + WS_CSUM;

    transpose_w_kernel<<<(3 * F_ * R_) / 256, 256, 0, stream>>>(W1, W2, Wg, wt);
    gemm3_kernel<<<NUM_GEMM_BLOCKS, 256, 0, stream>>>(x, wt, out, partials);
    reduce_kernel<<<1, 256, 0, stream>>>(partials, bias, gsum, csum);
    epilogue_kernel<<<(B_ * N_ * R_ / 4) / 256, 256, 0, stream>>>(out, gsum, csum);
}